// Graph_Inception_17532056502592
// MI455X (gfx1250) — compile-verified
//
#include <hip/hip_runtime.h>
#include <cstddef>
#include <cstdint>

typedef float v2f __attribute__((ext_vector_type(2)));
typedef float v8f __attribute__((ext_vector_type(8)));
typedef unsigned int u32x4 __attribute__((ext_vector_type(4)));
typedef int i32x4 __attribute__((ext_vector_type(4)));
typedef int i32x8 __attribute__((ext_vector_type(8)));

#define KC 64
#define BM 64
#define BN 32
#define LDS_K 66    // K-major row stride (floats): bank-conflict free, 8B-aligned pairs
#define GEMM_FLAG_BIAS 1
#define GEMM_FLAG_RELU 2

// Tensor Data Mover availability (device pass only; host pass parses fallback).
#if defined(__gfx1250__) && __has_builtin(__builtin_amdgcn_tensor_load_to_lds) && \
    __has_builtin(__builtin_amdgcn_s_wait_tensorcnt)
#define CDNA5_TDM 1
#endif
#if defined(__has_include)
#if __has_include(<hip/amd_detail/amd_gfx1250_TDM.h>)
#define CDNA5_TDM_6ARG 1   // therock-10.0 headers -> 6-arg builtin
#endif
#endif

// ---------------------------------------------------------------------------
// Generic fp32 WMMA GEMM:  C[M x Npad] = A[M x K] * B[K x N] (+bias)(ReLU)
// Block: 128 threads = 4 waves. Block tile: 64(M) x 32(N); each wave owns a
// 16x32 strip (two 16x16 WMMA tiles sharing the A fragment). K staged through
// LDS in chunks of 64: A tile via TDM (zero-fill OOB + hardware LDS padding
// to stride 66), B tile staged TRANSPOSED (K-major) so every fragment is one
// contiguous ds_load_b64. V_WMMA_F32_16X16X4_F32 core.
// grid.z = batch with element strides sA/sB/sC.
// ---------------------------------------------------------------------------
__global__ __launch_bounds__(128)
void gemm_wmma_f32_kernel(const float* __restrict__ A, const float* __restrict__ B,
                          const float* __restrict__ bias, float* __restrict__ C,
                          int M, int N, int K, int lda, int ldb, int ldc,
                          long long sA, long long sB, long long sC, int flags)
{
    __shared__ float As[BM][LDS_K];   // As[m][k]
    __shared__ float Bs[BN][LDS_K];   // Bs[n][k]  (transposed: K-major)

    const int tid  = threadIdx.x;
    const int lane = tid & 31;         // wave32
    const int wave = tid >> 5;
    const int half = lane >> 4;        // lanes 16-31: K+2 / M+8 halves
    const int mr   = lane & 15;
    const int nc   = lane & 15;

    const long long zb = blockIdx.z;
    A += zb * sA;  B += zb * sB;  C += zb * sC;

    const int m0 = blockIdx.y * BM;
    const int n0 = blockIdx.x * BN;

    v8f acc0 = {0.f,0.f,0.f,0.f,0.f,0.f,0.f,0.f};
    v8f acc1 = {0.f,0.f,0.f,0.f,0.f,0.f,0.f,0.f};

    for (int k0 = 0; k0 < K; k0 += KC) {
#if defined(CDNA5_TDM)
        if (wave == 0) {
            // ---- TDM descriptor: 2D tile 64(K-dw) x 64(rows), fp32 ----
            unsigned long long ga = (unsigned long long)(uintptr_t)A +
                4ull * ((unsigned long long)(unsigned)m0 * (unsigned)lda + (unsigned)k0);
            unsigned ldsa = (unsigned)(uintptr_t)(&As[0][0]);   // low 32b of LDS aperture addr
            unsigned td0  = (unsigned)(K - k0);                 // remaining width -> OOB reads = 0
            unsigned td1  = (unsigned)(M - m0);                 // remaining rows (>= 64 always)
            u32x4 g0;
            g0.x = 1u;                                           // count = 1 valid descriptor
            g0.y = ldsa;                                         // lds_addr
            g0.z = (unsigned)(ga & 0xffffffffu);                 // global_addr[31:0]
            g0.w = ((unsigned)(ga >> 32) & 0x01ffffffu) | (2u << 30);  // addr[56:32] | type=2
            i32x8 g1;
            g1[0] = (int)((2u << 16)        // data_size = 4 bytes
                        | (1u << 20)        // pad_enable
                        | (5u << 22)        // pad_interval: 64 dwords
                        | (1u << 25));      // pad_amount: 2 dwords -> LDS stride 66
            g1[1] = (int)((td0 & 0xffffu) << 16);                       // tensor_dim0[15:0]
            g1[2] = (int)((td0 >> 16) | ((td1 & 0xffffu) << 16));       // dim0[31:16] | dim1[15:0]
            g1[3] = (int)((td1 >> 16) | (64u << 16));                   // dim1[31:16] | tile_dim0=64
            g1[4] = (int)64u;                                           // tile_dim1=64, tile_dim2=0
            g1[5] = (int)(unsigned)lda;                                 // tensor_dim0_stride lo32
            g1[6] = 0;                                                  // stride hi16 | dim1_stride lo
            g1[7] = 0;
            i32x4 z4 = {0, 0, 0, 0};
#if defined(CDNA5_TDM_6ARG)
            i32x8 z8 = {0, 0, 0, 0, 0, 0, 0, 0};
            __builtin_amdgcn_tensor_load_to_lds(g0, g1, z4, z4, z8, 0);
#else
            __builtin_amdgcn_tensor_load_to_lds(g0, g1, z4, z4, 0);
#endif
            __builtin_amdgcn_s_wait_tensorcnt(0);
        }
#else
        // Fallback: cooperative A-tile stage (zero-fill past K)
        for (int idx = tid; idx < BM * KC; idx += 128) {
            int r = idx >> 6;
            int c = idx & (KC - 1);
            int gk = k0 + c;
            As[r][c] = (gk < K) ? A[(long long)(m0 + r) * lda + gk] : 0.f;
        }
#endif
        // B tile 64(K) x 32(N), stored transposed Bs[n][k].
        // gn varies fastest across tid -> coalesced global reads; LDS writes
        // at stride 66 floats land on 32 distinct banks.
        for (int idx = tid; idx < KC * BN; idx += 128) {
            int r = idx >> 5;           // k within chunk
            int c = idx & (BN - 1);     // n within tile
            int gk = k0 + r, gn = n0 + c;
            Bs[c][r] = (gk < K && gn < N) ? B[(long long)gk * ldb + gn] : 0.f;
        }
        __syncthreads();

        const int ra = wave * 16 + mr;
        #pragma unroll
        for (int kk = 0; kk < KC; kk += 4) {
            v2f a, b0, b1;
            // A 16x4 frag: lane holds row M=mr; VGPR v -> K = kk + 2*half + v
            a.x  = As[ra][kk + 2 * half + 0];
            a.y  = As[ra][kk + 2 * half + 1];
            // B 4x16 frags: contiguous K-pairs in transposed Bs -> ds_load_b64
            b0.x = Bs[nc][kk + 2 * half + 0];
            b0.y = Bs[nc][kk + 2 * half + 1];
            b1.x = Bs[nc + 16][kk + 2 * half + 0];
            b1.y = Bs[nc + 16][kk + 2 * half + 1];
            acc0 = __builtin_amdgcn_wmma_f32_16x16x4_f32(false, a, false, b0, (short)0, acc0, false, false);
            acc1 = __builtin_amdgcn_wmma_f32_16x16x4_f32(false, a, false, b1, (short)0, acc1, false, false);
        }
        __syncthreads();
    }

    // Epilogue: D layout: VGPR v -> row = v + 8*half, col = lane&15
    const int gn0 = n0 + nc;
    const int gn1 = gn0 + 16;
    const bool in0 = gn0 < N, in1 = gn1 < N;
    const float bv0 = ((flags & GEMM_FLAG_BIAS) && in0) ? bias[gn0] : 0.f;
    const float bv1 = ((flags & GEMM_FLAG_BIAS) && in1) ? bias[gn1] : 0.f;
    #pragma unroll
    for (int v = 0; v < 8; ++v) {
        const int row = m0 + wave * 16 + v + 8 * half;
        float x0 = acc0[v] + bv0;
        float x1 = acc1[v] + bv1;
        if (flags & GEMM_FLAG_RELU) { x0 = fmaxf(x0, 0.f); x1 = fmaxf(x1, 0.f); }
        C[(long long)row * ldc + gn0] = in0 ? x0 : 0.f;
        C[(long long)row * ldc + gn1] = in1 ? x1 : 0.f;
    }
}

// ---------------------------------------------------------------------------
// Per-column mean + rsqrt(var+eps) over M rows (population variance).
// ---------------------------------------------------------------------------
__global__ __launch_bounds__(256)
void colstats_kernel(const float* __restrict__ x, int ld, int M,
                     float* __restrict__ mean, float* __restrict__ rstd)
{
    __shared__ float ss[256], ss2[256];
    const int c = blockIdx.x;
    float s = 0.f, s2 = 0.f;
    for (int r = threadIdx.x; r < M; r += 256) {
        float v = x[(long long)r * ld + c];
        s += v;  s2 += v * v;
    }
    ss[threadIdx.x] = s;  ss2[threadIdx.x] = s2;
    __syncthreads();
    for (int o = 128; o > 0; o >>= 1) {
        if (threadIdx.x < o) {
            ss[threadIdx.x]  += ss[threadIdx.x + o];
            ss2[threadIdx.x] += ss2[threadIdx.x + o];
        }
        __syncthreads();
    }
    if (threadIdx.x == 0) {
        float mu  = ss[0] / (float)M;
        float var = ss2[0] / (float)M - mu * mu;
        mean[c] = mu;
        rstd[c] = rsqrtf(var + 1e-5f);
    }
}

// Per-column min over M rows (dummy row for neighbor max-pool).
__global__ __launch_bounds__(256)
void colmin_kernel(const float* __restrict__ x, int ld, int M, float* __restrict__ cmin)
{
    __shared__ float sm[256];
    const int c = blockIdx.x;
    float m = 3.402823466e+38f;
    for (int r = threadIdx.x; r < M; r += 256)
        m = fminf(m, x[(long long)r * ld + c]);
    sm[threadIdx.x] = m;
    __syncthreads();
    for (int o = 128; o > 0; o >>= 1) {
        if (threadIdx.x < o) sm[threadIdx.x] = fminf(sm[threadIdx.x], sm[threadIdx.x + o]);
        __syncthreads();
    }
    if (threadIdx.x == 0) cmin[c] = sm[0];
}

// y = relu?( gamma*(x-mean)*rstd + beta ), written into a column slice of out.
__global__ __launch_bounds__(256)
void bn_apply_kernel(const float* __restrict__ x, int ldx,
                     const float* __restrict__ mean, const float* __restrict__ rstd,
                     const float* __restrict__ gamma, const float* __restrict__ beta,
                     float* __restrict__ out, int ldo, int off,
                     long long M, int nc, int relu)
{
    const long long total = M * (long long)nc;
    for (long long i = (long long)blockIdx.x * 256 + threadIdx.x; i < total;
         i += (long long)gridDim.x * 256) {
        long long r = i / nc;
        int c = (int)(i - r * nc);
        float v = (x[r * ldx + c] - mean[c]) * rstd[c];
        if (gamma) v = v * gamma[c] + beta[c];
        if (relu)  v = fmaxf(v, 0.f);
        out[r * ldo + off + c] = v;
    }
}

// Neighbor max-pool: out[i][c] = max_j h[pnl[i][j]][c], index==16384 -> column min.
__global__ __launch_bounds__(128)
void maxpool_kernel(const float* __restrict__ h, int ldh,
                    const int* __restrict__ pnl, const float* __restrict__ cmin,
                    float* __restrict__ out, int ldo, int off, int d)
{
    __shared__ int idx[16];
    const int row = blockIdx.x;
    if (threadIdx.x < 16) idx[threadIdx.x] = pnl[row * 16 + threadIdx.x];
    __syncthreads();
    for (int c = threadIdx.x; c < d; c += 128) {
        float m = -3.402823466e+38f;
        #pragma unroll
        for (int j = 0; j < 16; ++j) {
            int id = idx[j];
            float v = (id == 16384) ? cmin[c] : h[(long long)id * ldh + c];
            m = fmaxf(m, v);
        }
        out[(long long)row * ldo + off + c] = m;
    }
}

// Final max/min/mean/sum pooling over N=2048 per batch.
__global__ __launch_bounds__(256)
void finalpool_kernel(const float* __restrict__ h, int ld, int d, float* __restrict__ out)
{
    __shared__ float smx[256], smn[256], ssm[256];
    const int c = blockIdx.x;
    const int b = blockIdx.y;
    const float* p = h + (long long)b * 2048 * ld + c;
    float mx = -3.402823466e+38f, mn = 3.402823466e+38f, sm = 0.f;
    for (int r = threadIdx.x; r < 2048; r += 256) {
        float v = p[(long long)r * ld];
        mx = fmaxf(mx, v);  mn = fminf(mn, v);  sm += v;
    }
    smx[threadIdx.x] = mx;  smn[threadIdx.x] = mn;  ssm[threadIdx.x] = sm;
    __syncthreads();
    for (int o = 128; o > 0; o >>= 1) {
        if (threadIdx.x < o) {
            smx[threadIdx.x] = fmaxf(smx[threadIdx.x], smx[threadIdx.x + o]);
            smn[threadIdx.x] = fminf(smn[threadIdx.x], smn[threadIdx.x + o]);
            ssm[threadIdx.x] += ssm[threadIdx.x + o];
        }
        __syncthreads();
    }
    if (threadIdx.x == 0) {
        out[(long long)b * 4 * d + 0 * d + c] = smx[0];
        out[(long long)b * 4 * d + 1 * d + c] = smn[0];
        out[(long long)b * 4 * d + 2 * d + c] = ssm[0] / 2048.f;
        out[(long long)b * 4 * d + 3 * d + c] = ssm[0];
    }
}

// ---------------------------------------------------------------------------
// Host side
// ---------------------------------------------------------------------------
struct GcnP { const float *W1, *b1, *W2, *b2, *gamma, *beta; int hid, dout; };

static GcnP read_gcn(void* const* d_in, const int* sz, int base, int hid, int dout)
{
    GcnP p; p.hid = hid; p.dout = dout;
    auto f = [&](int i) { return (const float*)d_in[i]; };
    if (sz[base + 1] == hid) {
        // insertion order: W1, b1, W2, b2, gamma, beta
        p.W1 = f(base);     p.b1 = f(base + 1); p.W2 = f(base + 2);
        p.b2 = f(base + 3); p.gamma = f(base + 4); p.beta = f(base + 5);
    } else {
        // jax sorted-key order: W1, W2, b1, b2, beta, gamma
        p.W1 = f(base);     p.W2 = f(base + 1); p.b1 = f(base + 2);
        p.b2 = f(base + 3); p.beta = f(base + 4); p.gamma = f(base + 5);
    }
    return p;
}

static void launch_gemm(hipStream_t s, const float* A, const float* B, const float* bias,
                        float* C, int M, int N, int K, int lda, int ldb, int ldc,
                        long long sA, long long sB, long long sC, int nb, int flags)
{
    dim3 g((N + BN - 1) / BN, M / BM, nb);
    gemm_wmma_f32_kernel<<<g, 128, 0, s>>>(A, B, bias, C, M, N, K, lda, ldb, ldc,
                                           sA, sB, sC, flags);
}

static void run_gcn(hipStream_t s, const float* agg, int ldagg, int din, const GcnP& g,
                    float* t1, float* x2, float* mean, float* rstd,
                    float* dst, int ldd, int off)
{
    // t1 = relu(agg @ W1 + b1)
    launch_gemm(s, agg, g.W1, g.b1, t1, 16384, g.hid, din, ldagg, g.hid, 128,
                0, 0, 0, 1, GEMM_FLAG_BIAS | GEMM_FLAG_RELU);
    // x2 = t1 @ W2 + b2
    launch_gemm(s, t1, g.W2, g.b2, x2, 16384, g.dout, g.hid, 128, g.dout, 128,
                0, 0, 0, 1, GEMM_FLAG_BIAS);
    // BN (batch stats) + ReLU into concat slice
    colstats_kernel<<<g.dout, 256, 0, s>>>(x2, 128, 16384, mean, rstd);
    long long total = 16384LL * g.dout;
    int blocks = (int)((total + 255) / 256);
    bn_apply_kernel<<<blocks, 256, 0, s>>>(x2, 128, mean, rstd, g.gamma, g.beta,
                                           dst, ldd, off, 16384, g.dout, 1);
}

extern "C" void kernel_launch(void* const* d_in, const int* in_sizes, int n_in,
                              void* d_out, int out_size, void* d_ws, size_t ws_size,
                              hipStream_t stream)
{
    if (n_in < 28) return;  // expect X, A, pnl, first_weight, 4 gcn param groups x6

    const float* X      = (const float*)d_in[0];   // (16384, 136)
    const float* Aadj   = (const float*)d_in[1];   // (2048, 2048)
    const int*   pnl    = (const int*)d_in[2];     // (16384, 16)
    const float* Wfirst = (const float*)d_in[3];   // (136, 136)

    GcnP g01 = read_gcn(d_in, in_sizes, 4,  66,  128);
    GcnP g02 = read_gcn(d_in, in_sizes, 10, 50,  64);
    GcnP g11 = read_gcn(d_in, in_sizes, 16, 114, 128);
    GcnP g12 = read_gcn(d_in, in_sizes, 22, 98,  64);

    // Workspace layout (floats); padded row strides (mult. of 32):
    // 136 -> 160, 328 -> 352, 520 -> 544
    float* ws   = (float*)d_ws;
    float* mean = ws;                 // 1024
    float* rstd = ws + 1024;          // 1024
    float* cmin = ws + 2048;          // 1024 (+ slack to 4096)
    float* xbn  = ws + 4096;                           // 16384*136
    float* h0   = xbn + 16384LL * 136;                 // 16384*160
    float* agg  = h0  + 16384LL * 160;                 // 16384*352 (shared by both layers)
    float* t1   = agg + 16384LL * 352;                 // 16384*128
    float* x2   = t1  + 16384LL * 128;                 // 16384*128
    float* h1   = x2  + 16384LL * 128;                 // 16384*352
    float* h2   = h1  + 16384LL * 352;                 // 16384*544

    // ---- bn0 (no affine) + first linear ----
    colstats_kernel<<<136, 256, 0, stream>>>(X, 136, 16384, mean, rstd);
    {
        long long total = 16384LL * 136;
        int blocks = (int)((total + 255) / 256);
        bn_apply_kernel<<<blocks, 256, 0, stream>>>(X, 136, mean, rstd, nullptr, nullptr,
                                                    xbn, 136, 0, 16384, 136, 0);
    }
    launch_gemm(stream, xbn, Wfirst, nullptr, h0, 16384, 136, 136, 136, 136, 160,
                0, 0, 0, 1, 0);

    // ---- Inception layer 0 (din = 136) ----
    // agg = A @ h0 per batch (identical for gcn1 and gcn2 -> compute once)
    launch_gemm(stream, Aadj, h0, nullptr, agg, 2048, 136, 2048, 2048, 160, 160,
                0, 2048LL * 160, 2048LL * 160, 8, 0);
    run_gcn(stream, agg, 160, 136, g01, t1, x2, mean, rstd, h1, 352, 136);
    run_gcn(stream, agg, 160, 136, g02, t1, x2, mean, rstd, h1, 352, 264);
    colmin_kernel<<<136, 256, 0, stream>>>(h0, 160, 16384, cmin);
    maxpool_kernel<<<16384, 128, 0, stream>>>(h0, 160, pnl, cmin, h1, 352, 0, 136);

    // ---- Inception layer 1 (din = 328) ----
    launch_gemm(stream, Aadj, h1, nullptr, agg, 2048, 328, 2048, 2048, 352, 352,
                0, 2048LL * 352, 2048LL * 352, 8, 0);
    run_gcn(stream, agg, 352, 328, g11, t1, x2, mean, rstd, h2, 544, 328);
    run_gcn(stream, agg, 352, 328, g12, t1, x2, mean, rstd, h2, 544, 456);
    colmin_kernel<<<328, 256, 0, stream>>>(h1, 352, 16384, cmin);
    maxpool_kernel<<<16384, 128, 0, stream>>>(h1, 352, pnl, cmin, h2, 544, 0, 328);

    // ---- Final max/min/mean/sum pooling over N per batch -> (8, 2080) ----
    finalpool_kernel<<<dim3(520, 8), 256, 0, stream>>>(h2, 544, 520, (float*)d_out);
}